// GraphFractalityPrior_85736137163440
// MI455X (gfx1250) — compile-verified
//
#include <hip/hip_runtime.h>

typedef float v2f __attribute__((ext_vector_type(2)));
typedef float v8f __attribute__((ext_vector_type(8)));

#define N 2048
#define HEADS 8
#define BATCH 4

// Exact wave32 sum via V_WMMA_F32_16X16X4_F32 (ones-matmul reduction).
// A layout (16x4 f32): VGPR0 lanes0-15 -> A[M=lane][K=0], lanes16-31 -> A[M=lane-16][K=2];
// VGPR1 -> K=1/K=3.  With A=(v,0) and B=all-ones: D[m][n] = v_m + v_{m+16}.
// Summing D's 8 VGPRs gives half-wave sums (uniform per 16-lane half); a second
// WMMA folds the halves -> total broadcast to all lanes. Values are small
// integers, so f32 arithmetic is exact.
__device__ __forceinline__ float wave_sum32_wmma(float v) {
    v2f a;    a[0] = v;    a[1] = 0.0f;
    v2f ones; ones[0] = 1.0f; ones[1] = 1.0f;
    v8f c = {};
    v8f d = __builtin_amdgcn_wmma_f32_16x16x4_f32(false, a, false, ones,
                                                  (short)0, c, false, false);
    float s = d[0] + d[1] + d[2] + d[3] + d[4] + d[5] + d[6] + d[7];
    v2f a2; a2[0] = s; a2[1] = 0.0f;
    v8f d2 = __builtin_amdgcn_wmma_f32_16x16x4_f32(false, a2, false, ones,
                                                   (short)0, c, false, false);
    return d2[0];
}

__global__ void frac_init(unsigned int* __restrict__ cnt) {
    if (threadIdx.x < 3 * BATCH) cnt[threadIdx.x] = 0u;
}

// Grid: (2, 256, 4) blocks of 256 threads.
//  z = batch, y = 8-row strip, x = 1024-col half.
// Each thread: 8 rows x 4 cols (one float4) x 8 heads = 1 KB streamed.
__global__ __launch_bounds__(256) void frac_count(const float* __restrict__ aw,
                                                  unsigned int* __restrict__ cnt) {
    const int t    = threadIdx.x;
    const int b    = blockIdx.z;
    const int row0 = blockIdx.y * 8;
    const int col0 = blockIdx.x * 1024 + t * 4;

    const size_t NNh = (size_t)N * (size_t)N;                     // per-head plane
    const float* p = aw + (size_t)b * HEADS * NNh + (size_t)row0 * N + col0;

    unsigned pm[4] = {0u, 0u, 0u, 0u};   // per row-pair: 4-bit column mask
    unsigned qm[2] = {0u, 0u};           // per row-quad: 4-bit column mask

    #pragma unroll
    for (int r = 0; r < 8; ++r) {
        const float* pr = p + (size_t)r * N;
        float4 s = *reinterpret_cast<const float4*>(pr);          // head 0
        #pragma unroll
        for (int h = 1; h < HEADS; ++h) {
            float4 v = *reinterpret_cast<const float4*>(pr + (size_t)h * NNh);
            s.x += v.x; s.y += v.y; s.z += v.z; s.w += v.w;
        }
        // mean over heads > 0.1  (mean = sum * 1/8, exact scaling)
        unsigned rm = (unsigned)(s.x * 0.125f > 0.1f)
                    | ((unsigned)(s.y * 0.125f > 0.1f) << 1)
                    | ((unsigned)(s.z * 0.125f > 0.1f) << 2)
                    | ((unsigned)(s.w * 0.125f > 0.1f) << 3);
        pm[r >> 1] |= rm;
        qm[r >> 2] |= rm;
    }

    // scale 2: 4 row-pairs x 2 col-pairs fully inside this strip
    int c2 = 0;
    #pragma unroll
    for (int i = 0; i < 4; ++i)
        c2 += (int)((pm[i] & 0x3u) != 0u) + (int)((pm[i] & 0xCu) != 0u);
    // scale 4: 2 row-quads x 1 col-quad fully inside this strip
    int c4 = (int)(qm[0] != 0u) + (int)(qm[1] != 0u);
    // scale 8: 8x8 box spans lane pair (t, t^1); cols are 8-aligned for even t
    int anyb = ((qm[0] | qm[1]) != 0u) ? 1 : 0;
    int nb   = __shfl_xor(anyb, 1, 32);
    int c8   = ((t & 1) == 0) ? (anyb | nb) : 0;

    // wave32 reduction through the matrix pipe (exact small-int f32 sums)
    float s2 = wave_sum32_wmma((float)c2);
    float s4 = wave_sum32_wmma((float)c4);
    float s8 = wave_sum32_wmma((float)c8);

    __shared__ unsigned int sc[3];
    if (t < 3) sc[t] = 0u;
    __syncthreads();
    if ((t & 31) == 0) {
        atomicAdd(&sc[0], (unsigned)(s2 + 0.5f));
        atomicAdd(&sc[1], (unsigned)(s4 + 0.5f));
        atomicAdd(&sc[2], (unsigned)(s8 + 0.5f));
    }
    __syncthreads();
    if (t == 0) {
        atomicAdd(&cnt[b * 3 + 0], sc[0]);
        atomicAdd(&cnt[b * 3 + 1], sc[1]);
        atomicAdd(&cnt[b * 3 + 2], sc[2]);
    }
}

// Log-log least-squares slope + loss. x = log(1/s) for s in {2,4,8};
// centered xc = {ln2, 0, -ln2}, sum(xc^2) = 2 ln2^2; y-centering cancels.
__global__ void frac_final(const unsigned int* __restrict__ cnt,
                           float* __restrict__ out) {
    if (threadIdx.x != 0) return;
    const float LN2   = 0.69314718055994531f;
    const float denom = 2.0f * LN2 * LN2;
    float acc = 0.0f;
    for (int b = 0; b < BATCH; ++b) {
        float y0 = __logf((float)cnt[b * 3 + 0] + 1.0f);   // s = 2
        float y2 = __logf((float)cnt[b * 3 + 2] + 1.0f);   // s = 8
        float dB = (LN2 * y0 - LN2 * y2) / denom;          // xc[1] = 0
        float e  = dB - 2.0f;                              // TARGET_DIM
        acc += e * e;
    }
    out[0] = 0.005f * (acc * 0.25f);                       // WEIGHT * mean
}

extern "C" void kernel_launch(void* const* d_in, const int* in_sizes, int n_in,
                              void* d_out, int out_size, void* d_ws, size_t ws_size,
                              hipStream_t stream) {
    const float* aw = (const float*)d_in[0];
    unsigned int* cnt = (unsigned int*)d_ws;   // 12 counters: [batch][scale]

    frac_init<<<1, 32, 0, stream>>>(cnt);
    frac_count<<<dim3(2, 256, BATCH), 256, 0, stream>>>(aw, cnt);
    frac_final<<<1, 1, 0, stream>>>(cnt, (float*)d_out);
}